// RepFlowLayerS_21672404976277
// MI455X (gfx1250) — compile-verified
//
#include <hip/hip_runtime.h>
#include <math.h>

// ---- problem constants (from reference) ----
#define NLOC    4096
#define N_EDGE  262144
#define N_ANGLE 524288
#define N_DIM   128
#define E_DIM   64
#define A_DIM   32
// DYN_E = 12, DYN_A = 4

// LDS row strides padded to (stride % 32) == 4 for conflict-free A reads
#define LDE 324   // edge tile, logical 320
#define LDA 292   // angle tile, logical 288
#define LDN 132   // node tile, logical 128
#define LDS_ 772  // sym tile, logical 768
#define LDR 68    // reduced tile, logical 64

typedef __attribute__((ext_vector_type(2))) float v2f;
typedef __attribute__((ext_vector_type(8))) float v8f;

// D = A(16x4) * B(4x16) + C(16x16), full f32 (matches reference precision).
__device__ __forceinline__ v8f wmma4(v2f a, v2f b, v8f c) {
    return __builtin_amdgcn_wmma_f32_16x16x4_f32(
        false, a, false, b, (short)0, c, false, false);
}

// x * rcp(1+exp(-x)): v_exp_f32 + v_rcp_f32, no IEEE divide sequence.
__device__ __forceinline__ float silu(float x) {
    float e = __expf(-x);
    return x * __builtin_amdgcn_rcpf(1.0f + e);
}

__device__ __forceinline__ void atomic_add_f32(float* p, float v) {
    unsafeAtomicAdd(p, v);   // global_atomic_add_f32 (no CAS loop)
}

// ---------------------------------------------------------------------------
// Stage 2: per-edge outer-product scatter for the symmetrization tensors.
// ---------------------------------------------------------------------------
__global__ __launch_bounds__(256) void sym_scatter(
    const float* __restrict__ node_ext, const float* __restrict__ edge_ebd,
    const float* __restrict__ h2, const float* __restrict__ sw,
    const int* __restrict__ n2e, const int* __restrict__ next2e,
    float* __restrict__ acc_e, float* __restrict__ acc_n)
{
    int e = blockIdx.x * 4 + (threadIdx.x >> 6);
    int d = threadIdx.x & 63;       // 0..63
    float s  = sw[e];
    int owner = n2e[e];
    int nei   = next2e[e];
    float h0 = h2[e*3+0], h1 = h2[e*3+1], hz = h2[e*3+2];

    float ev = edge_ebd[e*E_DIM + d] * s;
    float* pe = acc_e + owner * (3*E_DIM);
    atomic_add_f32(pe + 0*E_DIM + d, h0*ev);
    atomic_add_f32(pe + 1*E_DIM + d, h1*ev);
    atomic_add_f32(pe + 2*E_DIM + d, hz*ev);

    float* pn = acc_n + owner * (3*N_DIM);
    #pragma unroll
    for (int j = 0; j < 2; ++j) {
        int dd = d + j*64;
        float nv = node_ext[nei*N_DIM + dd] * s;
        atomic_add_f32(pn + 0*N_DIM + dd, h0*nv);
        atomic_add_f32(pn + 1*N_DIM + dd, h1*nv);
        atomic_add_f32(pn + 2*N_DIM + dd, hz*nv);
    }
}

// ---------------------------------------------------------------------------
// Stage 3: g = (h2g2[:, :4]^T @ h2g2) / 36   ( = DYN_E^-1 * 1/3 )
// sym_cat layout: [atom][ 4*64 edge part | 4*128 node part ] = 768 wide
// ---------------------------------------------------------------------------
__global__ __launch_bounds__(192) void sym_finalize(
    const float* __restrict__ acc_e, const float* __restrict__ acc_n,
    float* __restrict__ sym_cat)
{
    int atom = blockIdx.x;
    int d = threadIdx.x;            // 0..191
    const float k36 = 1.0f / 36.0f;
    if (d < E_DIM) {
        const float* p = acc_e + atom * (3*E_DIM);
        float v0 = p[d], v1 = p[E_DIM + d], v2 = p[2*E_DIM + d];
        float* out = sym_cat + atom*768;
        #pragma unroll
        for (int a = 0; a < 4; ++a) {
            float g = (p[a]*v0 + p[E_DIM + a]*v1 + p[2*E_DIM + a]*v2) * k36;
            out[a*E_DIM + d] = g;
        }
    } else {
        int dd = d - E_DIM;
        const float* p = acc_n + atom * (3*N_DIM);
        float v0 = p[dd], v1 = p[N_DIM + dd], v2 = p[2*N_DIM + dd];
        float* out = sym_cat + atom*768 + 4*E_DIM;
        #pragma unroll
        for (int a = 0; a < 4; ++a) {
            float g = (p[a]*v0 + p[N_DIM + a]*v1 + p[2*N_DIM + a]*v2) * k36;
            out[a*N_DIM + dd] = g;
        }
    }
}

// ---------------------------------------------------------------------------
// Stage 4: edge GEMM. edge_info (K=320) x [W_ne | W_es] (N=192), fused gather.
// 1 block = 16 edges; 4 waves x 3 N-tiles; K loop of 80 WMMA f32 16x16x4.
// The (col < 128) split is tile-aligned, so each wave takes one side
// uniformly: EXEC is all-ones around every WMMA, and each side indexes its
// weight array with a uniform base (saddr + 32-bit scaled offset).
// ---------------------------------------------------------------------------
__global__ __launch_bounds__(128) void edge_gemm(
    const float* __restrict__ node_ext, const float* __restrict__ edge_ebd,
    const float* __restrict__ sw,
    const int* __restrict__ n2e, const int* __restrict__ next2e,
    const float* __restrict__ W_ne, const float* __restrict__ b_ne,
    const float* __restrict__ W_es, const float* __restrict__ b_es,
    const float* __restrict__ r_e0,
    float* __restrict__ node_msg, float* __restrict__ e_out)
{
    __shared__ float tile[16 * LDE];
    __shared__ int   s_i0[16], s_i1[16];
    int m0  = blockIdx.x * 16;
    int tid = threadIdx.x;

    if (tid < 16) {
        s_i0[tid] = n2e[m0 + tid];
        s_i1[tid] = next2e[m0 + tid];
    }
    __syncthreads();

    for (int idx = tid; idx < 16*320; idx += 128) {
        int r = idx / 320, c = idx % 320;
        float v;
        if (c < 128)      v = node_ext[s_i0[r]*N_DIM + c];
        else if (c < 256) v = node_ext[s_i1[r]*N_DIM + (c-128)];
        else              v = edge_ebd[(m0 + r)*E_DIM + (c-256)];
        tile[r*LDE + c] = v;
    }
    __syncthreads();

    int lane = tid & 31, wid = tid >> 5;
    int half = lane >> 4, lr = lane & 15;
    int rbase = half * 8;

    for (int t = wid; t < 12; t += 4) {          // wave-uniform
        int n0  = t * 16;
        int col = n0 + lr;
        if (col < 128) {                         // uniform per wave (tile-aligned)
            v8f acc = {};
            #pragma unroll 4
            for (int k = 0; k < 320; k += 4) {
                int ka = k + 2*half;
                v2f a, b;
                a.x = tile[lr*LDE + ka];
                a.y = tile[lr*LDE + ka + 1];
                b.x = W_ne[ka*N_DIM + col];
                b.y = W_ne[(ka+1)*N_DIM + col];
                acc = wmma4(a, b, acc);
            }
            float bias = b_ne[col];
            #pragma unroll
            for (int v = 0; v < 8; ++v) {
                int row = m0 + rbase + v;        // edge id
                float x = silu(acc[v] + bias);
                atomic_add_f32(&node_msg[s_i0[rbase+v]*N_DIM + col], x * sw[row]);
            }
        } else {
            int c2 = col - 128;
            v8f acc = {};
            #pragma unroll 4
            for (int k = 0; k < 320; k += 4) {
                int ka = k + 2*half;
                v2f a, b;
                a.x = tile[lr*LDE + ka];
                a.y = tile[lr*LDE + ka + 1];
                b.x = W_es[ka*E_DIM + c2];
                b.y = W_es[(ka+1)*E_DIM + c2];
                acc = wmma4(a, b, acc);
            }
            float bias = b_es[c2];
            #pragma unroll
            for (int v = 0; v < 8; ++v) {
                int row = m0 + rbase + v;
                float x = silu(acc[v] + bias);
                e_out[row*E_DIM + c2] = edge_ebd[row*E_DIM + c2] + r_e0[c2] * x;
            }
        }
    }
}

// ---------------------------------------------------------------------------
// Stage 5: angle GEMM. angle_info (K=288) x [W_ea1 | W_as] (N=96).
// ---------------------------------------------------------------------------
__global__ __launch_bounds__(128) void angle_gemm(
    const float* __restrict__ node_ext, const float* __restrict__ edge_ebd,
    const float* __restrict__ angle_ebd, const float* __restrict__ a_sw,
    const int* __restrict__ n2a, const int* __restrict__ eij,
    const int* __restrict__ eik,
    const float* __restrict__ W_ea1, const float* __restrict__ b_ea1,
    const float* __restrict__ W_as,  const float* __restrict__ b_as,
    const float* __restrict__ r_a0,
    float* __restrict__ reduced, float* __restrict__ a_out)
{
    __shared__ float tile[16 * LDA];
    __shared__ int   s_n[16], s_ij[16], s_ik[16];
    int m0  = blockIdx.x * 16;
    int tid = threadIdx.x;

    if (tid < 16) {
        s_n[tid]  = n2a[m0 + tid];
        s_ij[tid] = eij[m0 + tid];
        s_ik[tid] = eik[m0 + tid];
    }
    __syncthreads();

    for (int idx = tid; idx < 16*288; idx += 128) {
        int r = idx / 288, c = idx % 288;
        float v;
        if (c < 32)       v = angle_ebd[(m0 + r)*A_DIM + c];
        else if (c < 160) v = node_ext[s_n[r]*N_DIM + (c-32)];
        else if (c < 224) v = edge_ebd[s_ik[r]*E_DIM + (c-160)];
        else              v = edge_ebd[s_ij[r]*E_DIM + (c-224)];
        tile[r*LDA + c] = v;
    }
    __syncthreads();

    int lane = tid & 31, wid = tid >> 5;
    int half = lane >> 4, lr = lane & 15;
    int rbase = half * 8;

    for (int t = wid; t < 6; t += 4) {           // wave-uniform
        int n0  = t * 16;
        int col = n0 + lr;
        if (col < 64) {                          // uniform per wave (tile-aligned)
            v8f acc = {};
            #pragma unroll 4
            for (int k = 0; k < 288; k += 4) {
                int ka = k + 2*half;
                v2f a, b;
                a.x = tile[lr*LDA + ka];
                a.y = tile[lr*LDA + ka + 1];
                b.x = W_ea1[ka*E_DIM + col];
                b.y = W_ea1[(ka+1)*E_DIM + col];
                acc = wmma4(a, b, acc);
            }
            float bias = b_ea1[col];
            #pragma unroll
            for (int v = 0; v < 8; ++v) {
                int row = m0 + rbase + v;        // angle id
                float x = silu(acc[v] + bias);
                atomic_add_f32(&reduced[s_ij[rbase+v]*E_DIM + col], x * a_sw[row]);
            }
        } else {
            int c2 = col - 64;
            v8f acc = {};
            #pragma unroll 4
            for (int k = 0; k < 288; k += 4) {
                int ka = k + 2*half;
                v2f a, b;
                a.x = tile[lr*LDA + ka];
                a.y = tile[lr*LDA + ka + 1];
                b.x = W_as[ka*A_DIM + c2];
                b.y = W_as[(ka+1)*A_DIM + c2];
                acc = wmma4(a, b, acc);
            }
            float bias = b_as[c2];
            #pragma unroll
            for (int v = 0; v < 8; ++v) {
                int row = m0 + rbase + v;
                float x = silu(acc[v] + bias);
                a_out[row*A_DIM + c2] = angle_ebd[row*A_DIM + c2] + r_a0[c2] * x;
            }
        }
    }
}

// ---------------------------------------------------------------------------
// Stage 6: (reduced * DYN_A^-0.5) @ W_ea2, silu, e_out += r_e1 * x
// ---------------------------------------------------------------------------
__global__ __launch_bounds__(128) void edge_angle_gemm(
    const float* __restrict__ reduced,
    const float* __restrict__ W_ea2, const float* __restrict__ b_ea2,
    const float* __restrict__ r_e1, float* __restrict__ e_out)
{
    __shared__ float tile[16 * LDR];
    int m0  = blockIdx.x * 16;
    int tid = threadIdx.x;
    for (int idx = tid; idx < 16*64; idx += 128) {
        int r = idx >> 6, c = idx & 63;
        tile[r*LDR + c] = reduced[(m0 + r)*E_DIM + c] * 0.5f;   // DYN_A^-0.5
    }
    __syncthreads();

    int lane = tid & 31, wid = tid >> 5;
    int half = lane >> 4, lr = lane & 15;
    int col = wid * 16 + lr;

    v8f acc = {};
    #pragma unroll 4
    for (int k = 0; k < 64; k += 4) {
        int ka = k + 2*half;
        v2f a, b;
        a.x = tile[lr*LDR + ka];
        a.y = tile[lr*LDR + ka + 1];
        b.x = W_ea2[ka*E_DIM + col];
        b.y = W_ea2[(ka+1)*E_DIM + col];
        acc = wmma4(a, b, acc);
    }
    int rbase = half * 8;
    float bias = b_ea2[col];
    float q = r_e1[col];
    #pragma unroll
    for (int v = 0; v < 8; ++v) {
        int row = m0 + rbase + v;
        float x = silu(acc[v] + bias);
        e_out[row*E_DIM + col] += q * x;
    }
}

// ---------------------------------------------------------------------------
// Stage 7: node update. acc1 = node_ebd @ W_ns (K=128), acc2 = sym_cat @ W_sym
// (K=768); n_out = node + r0*silu(acc1+b) + r1*silu(acc2+b) + r2*msg/12.
// ---------------------------------------------------------------------------
__global__ __launch_bounds__(256) void node_gemm(
    const float* __restrict__ node_ext, const float* __restrict__ sym_cat,
    const float* __restrict__ node_msg,
    const float* __restrict__ W_ns,  const float* __restrict__ b_ns,
    const float* __restrict__ W_sym, const float* __restrict__ b_sym,
    const float* __restrict__ r_n0, const float* __restrict__ r_n1,
    const float* __restrict__ r_n2, float* __restrict__ n_out)
{
    __shared__ float tn[16 * LDN];
    __shared__ float ts[16 * LDS_];
    int m0  = blockIdx.x * 16;
    int tid = threadIdx.x;
    for (int idx = tid; idx < 16*128; idx += 256) {
        int r = idx >> 7, c = idx & 127;
        tn[r*LDN + c] = node_ext[(m0 + r)*N_DIM + c];
    }
    for (int idx = tid; idx < 16*768; idx += 256) {
        int r = idx / 768, c = idx % 768;
        ts[r*LDS_ + c] = sym_cat[(m0 + r)*768 + c];
    }
    __syncthreads();

    int lane = tid & 31, wid = tid >> 5;          // 8 waves, 8 N-tiles
    int half = lane >> 4, lr = lane & 15;
    int col = wid * 16 + lr;

    v8f acc1 = {};
    #pragma unroll 4
    for (int k = 0; k < 128; k += 4) {
        int ka = k + 2*half;
        v2f a, b;
        a.x = tn[lr*LDN + ka];
        a.y = tn[lr*LDN + ka + 1];
        b.x = W_ns[ka*N_DIM + col];
        b.y = W_ns[(ka+1)*N_DIM + col];
        acc1 = wmma4(a, b, acc1);
    }
    v8f acc2 = {};
    #pragma unroll 4
    for (int k = 0; k < 768; k += 4) {
        int ka = k + 2*half;
        v2f a, b;
        a.x = ts[lr*LDS_ + ka];
        a.y = ts[lr*LDS_ + ka + 1];
        b.x = W_sym[ka*N_DIM + col];
        b.y = W_sym[(ka+1)*N_DIM + col];
        acc2 = wmma4(a, b, acc2);
    }

    int rbase = half * 8;
    float bs = b_ns[col], by = b_sym[col];
    float q0 = r_n0[col], q1 = r_n1[col], q2 = r_n2[col];
    #pragma unroll
    for (int v = 0; v < 8; ++v) {
        int row = m0 + rbase + v;
        float x1 = silu(acc1[v] + bs);
        float x2 = silu(acc2[v] + by);
        float msg = node_msg[row*N_DIM + col] * (1.0f / 12.0f);  // / DYN_E
        n_out[row*N_DIM + col] =
            tn[(rbase+v)*LDN + col] + q0*x1 + q1*x2 + q2*msg;
    }
}

// ---------------------------------------------------------------------------
extern "C" void kernel_launch(void* const* d_in, const int* in_sizes, int n_in,
                              void* d_out, int out_size, void* d_ws, size_t ws_size,
                              hipStream_t stream) {
    const float* node_ext  = (const float*)d_in[0];
    const float* edge_ebd  = (const float*)d_in[1];
    const float* h2        = (const float*)d_in[2];
    const float* angle_ebd = (const float*)d_in[3];
    const float* sw        = (const float*)d_in[6];
    const float* a_sw      = (const float*)d_in[9];
    const int*   edge_index  = (const int*)d_in[10];
    const int*   angle_index = (const int*)d_in[11];
    const float* W_ns  = (const float*)d_in[12];
    const float* b_ns  = (const float*)d_in[13];
    const float* W_sym = (const float*)d_in[14];
    const float* b_sym = (const float*)d_in[15];
    const float* W_ne  = (const float*)d_in[16];
    const float* b_ne  = (const float*)d_in[17];
    const float* W_es  = (const float*)d_in[18];
    const float* b_es  = (const float*)d_in[19];
    const float* W_ea1 = (const float*)d_in[20];
    const float* b_ea1 = (const float*)d_in[21];
    const float* W_ea2 = (const float*)d_in[22];
    const float* b_ea2 = (const float*)d_in[23];
    const float* W_as  = (const float*)d_in[24];
    const float* b_as  = (const float*)d_in[25];
    const float* r_n0  = (const float*)d_in[26];
    const float* r_n1  = (const float*)d_in[27];
    const float* r_n2  = (const float*)d_in[28];
    const float* r_e0  = (const float*)d_in[29];
    const float* r_e1  = (const float*)d_in[30];
    const float* r_a0  = (const float*)d_in[31];

    const int* n2e    = edge_index;
    const int* next2e = edge_index + N_EDGE;
    const int* n2a    = angle_index;
    const int* eij    = angle_index + N_ANGLE;
    const int* eik    = angle_index + 2*N_ANGLE;

    // workspace layout (floats). Accumulators first so one memset covers them.
    float* ws       = (float*)d_ws;
    float* acc_e    = ws;                              // 4096*192
    float* acc_n    = acc_e + (size_t)NLOC*3*E_DIM;    // 4096*384
    float* node_msg = acc_n + (size_t)NLOC*3*N_DIM;    // 4096*128
    float* reduced  = node_msg + (size_t)NLOC*N_DIM;   // 262144*64
    float* sym_cat  = reduced + (size_t)N_EDGE*E_DIM;  // 4096*768 (no zero needed)
    size_t zero_bytes = ((size_t)NLOC*3*E_DIM + (size_t)NLOC*3*N_DIM +
                         (size_t)NLOC*N_DIM + (size_t)N_EDGE*E_DIM) * sizeof(float);

    float* n_out = (float*)d_out;                      // 4096*128
    float* e_out = n_out + (size_t)NLOC*N_DIM;         // 262144*64
    float* a_out = e_out + (size_t)N_EDGE*E_DIM;       // 524288*32

    hipMemsetAsync(d_ws, 0, zero_bytes, stream);

    sym_scatter<<<N_EDGE/4, 256, 0, stream>>>(node_ext, edge_ebd, h2, sw,
                                              n2e, next2e, acc_e, acc_n);
    sym_finalize<<<NLOC, 192, 0, stream>>>(acc_e, acc_n, sym_cat);
    edge_gemm<<<N_EDGE/16, 128, 0, stream>>>(node_ext, edge_ebd, sw, n2e, next2e,
                                             W_ne, b_ne, W_es, b_es, r_e0,
                                             node_msg, e_out);
    angle_gemm<<<N_ANGLE/16, 128, 0, stream>>>(node_ext, edge_ebd, angle_ebd, a_sw,
                                               n2a, eij, eik,
                                               W_ea1, b_ea1, W_as, b_as, r_a0,
                                               reduced, a_out);
    edge_angle_gemm<<<N_EDGE/16, 128, 0, stream>>>(reduced, W_ea2, b_ea2, r_e1, e_out);
    node_gemm<<<NLOC/16, 256, 0, stream>>>(node_ext, sym_cat, node_msg,
                                           W_ns, b_ns, W_sym, b_sym,
                                           r_n0, r_n1, r_n2, n_out);
}